// DIEN_51024211476558
// MI455X (gfx1250) — compile-verified
//
#include <hip/hip_runtime.h>
#include <math.h>

typedef __attribute__((ext_vector_type(2))) float v2f;
typedef __attribute__((ext_vector_type(8))) float v8f;

constexpr int NB = 256;   // batch
constexpr int LL = 200;   // history length
constexpr int DD = 128;   // hidden dim

__device__ __forceinline__ float sigm_(float x) { return 1.0f / (1.0f + expf(-x)); }
__device__ __forceinline__ float softplus_(float x) {
  // log(1 + e^x), numerically stable
  return (x > 20.0f) ? x : log1pf(expf(x));
}

// D = A(16x4,f32) * B(4x16,f32) + C(16x16,f32)
__device__ __forceinline__ v8f wmma4(v2f a, v2f b, v8f c) {
  return __builtin_amdgcn_wmma_f32_16x16x4_f32(false, a, false, b, (short)0, c,
                                               false, false);
}

// ---------------------------------------------------------------------------
// K1: embedding gather with max_norm=1 renorm. One wave per row.
// ---------------------------------------------------------------------------
__global__ __launch_bounds__(256) void embed_renorm(
    const float* __restrict__ emb, const int* __restrict__ idx,
    float* __restrict__ out, int nrows) {
  int wave = (blockIdx.x * blockDim.x + threadIdx.x) >> 5;
  int lane = threadIdx.x & 31;
  if (wave >= nrows) return;
  const float* src = emb + (long)idx[wave] * DD;
  float4 v = ((const float4*)src)[lane];
  float ss = v.x * v.x + v.y * v.y + v.z * v.z + v.w * v.w;
  for (int o = 16; o > 0; o >>= 1) ss += __shfl_xor(ss, o, 32);
  float n = sqrtf(ss);
  float s = fminf(1.0f, 1.0f / fmaxf(n, 1e-7f));
  v.x *= s; v.y *= s; v.z *= s; v.w *= s;
  ((float4*)(out + (long)wave * DD))[lane] = v;
}

// ---------------------------------------------------------------------------
// K2: GRU scan (torch gate order r,z,n). 16 blocks x 16 batch rows.
// 24 waves: wave = gate*8 + dblk. Hidden state in LDS, 200 sequential steps.
// gates: r = sig(xWr+hWr+b), z = sig(...), n = tanh(gi_n + r*gh_n), separate
// x/h accumulators because r multiplies only the hidden part of n.
// ---------------------------------------------------------------------------
__global__ __launch_bounds__(768) void gru_scan(
    const float* __restrict__ X,      // item_embs [B][L][D]
    const float* __restrict__ w_ih, const float* __restrict__ w_hh,
    const float* __restrict__ b_ih, const float* __restrict__ b_hh,
    float* __restrict__ outs) {      // [B][L][D]
  __shared__ float h_lds[16 * DD];
  __shared__ float rbuf[16 * DD];
  __shared__ float zbuf[16 * DD];
  const int tid = threadIdx.x;
  const int wave = tid >> 5, lane = tid & 31;
  const int gate = wave >> 3, dblk = wave & 7;
  const int hl = lane >> 4, lm = lane & 15;
  const int m0 = blockIdx.x * 16;
  const int c = dblk * 16 + lm;                 // column within D block

  for (int i = tid; i < 16 * DD; i += blockDim.x) h_lds[i] = 0.0f;
  __syncthreads();

  const float bic = b_ih[gate * DD + c];
  const float bhc = b_hh[gate * DD + c];
  // A row base: row = m0+lm; per-lane k offset 2*hl
  const float* Xrow = X + ((long)(m0 + lm) * LL) * DD + 2 * hl;
  // B: B[k][n] = w[(gate*D + c) * D + k]  (x @ W^T), contiguous k pair
  const float* BxP = w_ih + (long)(gate * DD + c) * DD + 2 * hl;
  const float* BhP = w_hh + (long)(gate * DD + c) * DD + 2 * hl;

  for (int t = 0; t < LL; ++t) {
    v8f accx = {0, 0, 0, 0, 0, 0, 0, 0};
    v8f acch = {0, 0, 0, 0, 0, 0, 0, 0};
    const float* xrt = Xrow + (long)t * DD;
#pragma unroll
    for (int kc = 0; kc < 32; ++kc) {
      const int k0 = kc * 4;
      v2f a_x = *(const v2f*)(xrt + k0);
      v2f a_h = *(const v2f*)(&h_lds[lm * DD + k0 + 2 * hl]);
      v2f b_x = *(const v2f*)(BxP + k0);
      v2f b_h = *(const v2f*)(BhP + k0);
      accx = wmma4(a_x, b_x, accx);
      acch = wmma4(a_h, b_h, acch);
    }
    if (gate < 2) {
      float* buf = (gate == 0) ? rbuf : zbuf;
#pragma unroll
      for (int i = 0; i < 8; ++i) {
        int mloc = i + 8 * hl;
        buf[mloc * DD + c] = sigm_(accx[i] + acch[i] + bic + bhc);
      }
    }
    __syncthreads();  // r,z visible; all reads of h_old complete
    if (gate == 2) {
#pragma unroll
      for (int i = 0; i < 8; ++i) {
        int mloc = i + 8 * hl;
        float r = rbuf[mloc * DD + c];
        float nn = tanhf(accx[i] + bic + r * (acch[i] + bhc));
        float z = zbuf[mloc * DD + c];
        float hold = h_lds[mloc * DD + c];
        float hnew = (1.0f - z) * nn + z * hold;
        h_lds[mloc * DD + c] = hnew;
        outs[((long)(m0 + mloc) * LL + t) * DD + c] = hnew;
      }
    }
    __syncthreads();  // h_new visible before next step's GEMM
  }
}

// ---------------------------------------------------------------------------
// K3a/K3b: auxiliary BCE loss, deterministic two-stage reduction.
// loss_i = y*min(softplus(-s),100) + (1-y)*min(softplus(s),100), s = outs.x
// ---------------------------------------------------------------------------
__global__ __launch_bounds__(256) void aux_partial(
    const float* __restrict__ outs, const float* __restrict__ X,
    const int* __restrict__ labels, float* __restrict__ partials) {
  __shared__ float wsum[8];
  int wave = threadIdx.x >> 5, lane = threadIdx.x & 31;
  int gw = blockIdx.x * 8 + wave;  // 0..799
  float acc = 0.0f;
  for (int row = gw; row < NB * LL; row += 800) {
    float4 a4 = ((const float4*)(outs + (long)row * DD))[lane];
    float4 b4 = ((const float4*)(X + (long)row * DD))[lane];
    float d = a4.x * b4.x + a4.y * b4.y + a4.z * b4.z + a4.w * b4.w;
    for (int o = 16; o > 0; o >>= 1) d += __shfl_xor(d, o, 32);
    if (lane == 0) {
      float y = (float)labels[row];
      acc += y * fminf(softplus_(-d), 100.0f) +
             (1.0f - y) * fminf(softplus_(d), 100.0f);
    }
  }
  if (lane == 0) wsum[wave] = acc;
  __syncthreads();
  if (threadIdx.x == 0) {
    float s = 0.0f;
    for (int i = 0; i < 8; ++i) s += wsum[i];
    partials[blockIdx.x] = s;
  }
}

__global__ void aux_final(const float* __restrict__ partials,
                          float* __restrict__ scalars) {
  if (threadIdx.x == 0) {
    float s = 0.0f;
    for (int i = 0; i < 100; ++i) s += partials[i];
    scalars[0] = s / (float)(NB * LL);
  }
}

// ---------------------------------------------------------------------------
// K4: AUGRU scan. Attention softmax over size-1 axis == 1.0 exactly, so
// u_hat = u and the Wa path vanishes. gates: u(0), r(1), h_hat(2).
// u = sig(x@Wu + h@Uu + bu); r = sig(x@Wr + h@Ur + br);
// h_hat = tanh(x@Wh + r*(h@Uh) + bh); h = (1-u)*h + u*h_hat.
// Weights used un-transposed: B[k][n] = W[k*D + n].
// ---------------------------------------------------------------------------
__global__ __launch_bounds__(768) void augru_scan(
    const float* __restrict__ X,  // outs [B][L][D]
    const float* __restrict__ Wu, const float* __restrict__ Uu, const float* __restrict__ bu,
    const float* __restrict__ Wr, const float* __restrict__ Ur, const float* __restrict__ br,
    const float* __restrict__ Wh, const float* __restrict__ Uh, const float* __restrict__ bh,
    const float* __restrict__ h0, float* __restrict__ hfin) {
  __shared__ float h_lds[16 * DD];
  __shared__ float ubuf[16 * DD];
  __shared__ float rbuf[16 * DD];
  const int tid = threadIdx.x;
  const int wave = tid >> 5, lane = tid & 31;
  const int gate = wave >> 3, dblk = wave & 7;
  const int hl = lane >> 4, lm = lane & 15;
  const int m0 = blockIdx.x * 16;
  const int c = dblk * 16 + lm;

  for (int i = tid; i < 16 * DD; i += blockDim.x)
    h_lds[i] = h0[(long)(m0 + i / DD) * DD + (i % DD)];
  __syncthreads();

  const float* Wx = (gate == 0) ? Wu : (gate == 1) ? Wr : Wh;
  const float* Ux = (gate == 0) ? Uu : (gate == 1) ? Ur : Uh;
  const float bc = ((gate == 0) ? bu : (gate == 1) ? br : bh)[c];
  const float* Xrow = X + ((long)(m0 + lm) * LL) * DD + 2 * hl;

  for (int t = 0; t < LL; ++t) {
    v8f accx = {0, 0, 0, 0, 0, 0, 0, 0};
    v8f acch = {0, 0, 0, 0, 0, 0, 0, 0};
    const float* xrt = Xrow + (long)t * DD;
#pragma unroll
    for (int kc = 0; kc < 32; ++kc) {
      const int k0 = kc * 4;
      const int k = k0 + 2 * hl;
      v2f a_x = *(const v2f*)(xrt + k0);
      v2f a_h = *(const v2f*)(&h_lds[lm * DD + k0 + 2 * hl]);
      v2f b_x = {Wx[(long)k * DD + c], Wx[(long)(k + 1) * DD + c]};
      v2f b_h = {Ux[(long)k * DD + c], Ux[(long)(k + 1) * DD + c]};
      accx = wmma4(a_x, b_x, accx);
      acch = wmma4(a_h, b_h, acch);
    }
    if (gate < 2) {
      float* buf = (gate == 0) ? ubuf : rbuf;
#pragma unroll
      for (int i = 0; i < 8; ++i) {
        int mloc = i + 8 * hl;
        buf[mloc * DD + c] = sigm_(accx[i] + acch[i] + bc);
      }
    }
    __syncthreads();
    if (gate == 2) {
#pragma unroll
      for (int i = 0; i < 8; ++i) {
        int mloc = i + 8 * hl;
        float r = rbuf[mloc * DD + c];
        float hhat = tanhf(accx[i] + r * acch[i] + bc);
        float u = ubuf[mloc * DD + c];
        float hold = h_lds[mloc * DD + c];
        h_lds[mloc * DD + c] = (1.0f - u) * hold + u * hhat;
      }
    }
    __syncthreads();
  }
  for (int i = tid; i < 16 * DD; i += blockDim.x)
    hfin[(long)(m0 + i / DD) * DD + (i % DD)] = h_lds[i];
}

// ---------------------------------------------------------------------------
// K5a: z1 = [h | item] @ W1 + b1  (M=256, N=128, K=256), WMMA.
// ---------------------------------------------------------------------------
__global__ __launch_bounds__(256) void dense1(
    const float* __restrict__ hf, const float* __restrict__ ti,
    const float* __restrict__ W1, const float* __restrict__ b1,
    float* __restrict__ z1) {
  const int wave = threadIdx.x >> 5, lane = threadIdx.x & 31;
  const int hl = lane >> 4, lm = lane & 15;
  const int m0 = blockIdx.x * 16;
  const int c = wave * 16 + lm;  // 0..127
  v8f acc = {0, 0, 0, 0, 0, 0, 0, 0};
  const float* A0 = hf + (long)(m0 + lm) * DD + 2 * hl;
  const float* A1 = ti + (long)(m0 + lm) * DD + 2 * hl;
#pragma unroll
  for (int kc = 0; kc < 64; ++kc) {
    const int k0 = kc * 4;
    const int k = k0 + 2 * hl;  // 0..255
    v2f a = (kc < 32) ? *(const v2f*)(A0 + k0) : *(const v2f*)(A1 + k0 - DD);
    v2f b = {W1[(long)k * DD + c], W1[(long)(k + 1) * DD + c]};
    acc = wmma4(a, b, acc);
  }
  float bc = b1[c];
#pragma unroll
  for (int i = 0; i < 8; ++i) {
    int mloc = i + 8 * hl;
    z1[(long)(m0 + mloc) * DD + c] = acc[i] + bc;
  }
}

// ---------------------------------------------------------------------------
// K5b: Dice (training-mode BatchNorm over batch, gamma=1,beta=0) in-place.
// ---------------------------------------------------------------------------
__global__ void dice_inplace(float* __restrict__ z, int ncols) {
  int c = threadIdx.x;
  if (c >= ncols) return;
  float s = 0.0f;
  for (int m = 0; m < NB; ++m) s += z[(long)m * ncols + c];
  float mu = s / (float)NB;
  float v = 0.0f;
  for (int m = 0; m < NB; ++m) {
    float d = z[(long)m * ncols + c] - mu;
    v += d * d;
  }
  float istd = 1.0f / sqrtf(v / (float)NB + 1e-5f);
  for (int m = 0; m < NB; ++m) {
    float x = z[(long)m * ncols + c];
    float p = sigm_((x - mu) * istd);
    z[(long)m * ncols + c] = p * x + (1.0f - p) * 0.1f * x;
  }
}

// ---------------------------------------------------------------------------
// K5c: z2 = z1d @ W2 + b2  (M=256, N=64, K=128), WMMA.
// ---------------------------------------------------------------------------
__global__ __launch_bounds__(128) void dense2(
    const float* __restrict__ z1, const float* __restrict__ W2,
    const float* __restrict__ b2, float* __restrict__ z2) {
  const int wave = threadIdx.x >> 5, lane = threadIdx.x & 31;
  const int hl = lane >> 4, lm = lane & 15;
  const int m0 = blockIdx.x * 16;
  const int c = wave * 16 + lm;  // 0..63
  v8f acc = {0, 0, 0, 0, 0, 0, 0, 0};
  const float* A0 = z1 + (long)(m0 + lm) * DD + 2 * hl;
#pragma unroll
  for (int kc = 0; kc < 32; ++kc) {
    const int k0 = kc * 4;
    const int k = k0 + 2 * hl;
    v2f a = *(const v2f*)(A0 + k0);
    v2f b = {W2[(long)k * 64 + c], W2[(long)(k + 1) * 64 + c]};
    acc = wmma4(a, b, acc);
  }
  float bc = b2[c];
#pragma unroll
  for (int i = 0; i < 8; ++i) {
    int mloc = i + 8 * hl;
    z2[(long)(m0 + mloc) * 64 + c] = acc[i] + bc;
  }
}

// ---------------------------------------------------------------------------
// K5d: Dice on z2 (stats in LDS), logit = sig(z2d @ Wf + bf), BCE over batch,
// out = 0.2 * aux + rec. Single block, deterministic tree reduce.
// ---------------------------------------------------------------------------
__global__ __launch_bounds__(256) void final_loss(
    const float* __restrict__ z2, const float* __restrict__ Wf,
    const float* __restrict__ bf, const int* __restrict__ ylab,
    const float* __restrict__ scalars, float* __restrict__ out) {
  __shared__ float mu_s[64], istd_s[64];
  __shared__ float red[256];
  const int t = threadIdx.x;
  if (t < 64) {
    float s = 0.0f;
    for (int m = 0; m < NB; ++m) s += z2[(long)m * 64 + t];
    float mu = s / (float)NB;
    float v = 0.0f;
    for (int m = 0; m < NB; ++m) {
      float d = z2[(long)m * 64 + t] - mu;
      v += d * d;
    }
    mu_s[t] = mu;
    istd_s[t] = 1.0f / sqrtf(v / (float)NB + 1e-5f);
  }
  __syncthreads();
  float sdot = 0.0f;
  for (int cc = 0; cc < 64; ++cc) {
    float x = z2[(long)t * 64 + cc];
    float p = sigm_((x - mu_s[cc]) * istd_s[cc]);
    float xd = p * x + (1.0f - p) * 0.1f * x;
    sdot += xd * Wf[cc];
  }
  float s = sdot + bf[0];
  float y = (float)ylab[t];
  red[t] = y * fminf(softplus_(-s), 100.0f) +
           (1.0f - y) * fminf(softplus_(s), 100.0f);
  __syncthreads();
  for (int st = 128; st > 0; st >>= 1) {
    if (t < st) red[t] += red[t + st];
    __syncthreads();
  }
  if (t == 0) out[0] = 0.2f * scalars[0] + red[0] / (float)NB;
}

// ---------------------------------------------------------------------------
extern "C" void kernel_launch(void* const* d_in, const int* in_sizes, int n_in,
                              void* d_out, int out_size, void* d_ws,
                              size_t ws_size, hipStream_t stream) {
  const float* emb   = (const float*)d_in[0];
  const float* w_ih  = (const float*)d_in[1];
  const float* w_hh  = (const float*)d_in[2];
  const float* b_ih  = (const float*)d_in[3];
  const float* b_hh  = (const float*)d_in[4];
  const float* Wu    = (const float*)d_in[5];
  const float* Uu    = (const float*)d_in[6];
  const float* bu    = (const float*)d_in[7];
  const float* Wr    = (const float*)d_in[8];
  const float* Ur    = (const float*)d_in[9];
  const float* br    = (const float*)d_in[10];
  const float* Wh    = (const float*)d_in[11];
  const float* Uh    = (const float*)d_in[12];
  const float* bh    = (const float*)d_in[13];
  // d_in[14] = Wa : unused — softmax over a size-1 axis is identically 1.
  const float* W1    = (const float*)d_in[15];
  const float* b1    = (const float*)d_in[16];
  const float* W2    = (const float*)d_in[17];
  const float* b2    = (const float*)d_in[18];
  const float* Wf    = (const float*)d_in[19];
  const float* bf    = (const float*)d_in[20];
  const float* h0    = (const float*)d_in[21];
  const int* hseq    = (const int*)d_in[22];
  const int* hlab    = (const int*)d_in[23];
  const int* titem_i = (const int*)d_in[24];
  const int* tlab    = (const int*)d_in[25];

  float* ws = (float*)d_ws;
  float* item_embs = ws;                          // 256*200*128
  float* outs      = item_embs + (long)NB * LL * DD;
  float* titem     = outs + (long)NB * LL * DD;   // 256*128
  float* hfin      = titem + NB * DD;             // 256*128
  float* z1        = hfin + NB * DD;              // 256*128
  float* z2        = z1 + NB * DD;                // 256*64
  float* partials  = z2 + NB * 64;                // 100
  float* scalars   = partials + 128;              // aux loss

  // 1. embeddings (history: 51200 rows, target: 256 rows), 8 rows/block
  embed_renorm<<<(NB * LL) / 8, 256, 0, stream>>>(emb, hseq, item_embs, NB * LL);
  embed_renorm<<<NB / 8, 256, 0, stream>>>(emb, titem_i, titem, NB);
  // 2. GRU interest extraction
  gru_scan<<<16, 768, 0, stream>>>(item_embs, w_ih, w_hh, b_ih, b_hh, outs);
  // 3. auxiliary loss
  aux_partial<<<100, 256, 0, stream>>>(outs, item_embs, hlab, partials);
  aux_final<<<1, 32, 0, stream>>>(partials, scalars);
  // 4. AUGRU interest evolution
  augru_scan<<<16, 768, 0, stream>>>(outs, Wu, Uu, bu, Wr, Ur, br, Wh, Uh, bh,
                                     h0, hfin);
  // 5. prediction head
  dense1<<<16, 256, 0, stream>>>(hfin, titem, W1, b1, z1);
  dice_inplace<<<1, 128, 0, stream>>>(z1, DD);
  dense2<<<16, 128, 0, stream>>>(z1, W2, b2, z2);
  final_loss<<<1, 256, 0, stream>>>(z2, Wf, bf, tlab, scalars, (float*)d_out);
}